// MessagePassing_26096221291259
// MI455X (gfx1250) — compile-verified
//
#include <hip/hip_runtime.h>

// out[dst[e]][:] += x[src[e]][:]   for e in [0, E), feature dim = 64, f32.
//
// Roofline: ~51 MFLOP vs ~400 MB of L2 traffic (x and out both L2-resident at
// 12.8 MB on a 192 MB L2). Bound by L2 atomic throughput -> no WMMA structure
// to exploit. gfx1250 paths used:
//   - global_load_async_to_lds_b128 (ASYNCcnt) to stage the int64 edge indices
//     into LDS once per block instead of 16x redundant per-lane loads
//   - s_wait_asynccnt + workgroup barrier for the async pipeline
//   - global_load_b128 coalesced row gathers (16 lanes x float4 = 256B row)
//   - non-returning relaxed agent-scope global_atomic_add_f32 (RMW in L2)
//   - global_prefetch_b8 run-ahead on the edge index stream

#define DIM 64
#define EDGES_PER_BLOCK 16    // 256 threads / 16 lanes-per-edge
#define THREADS 256

__global__ void __launch_bounds__(THREADS)
mp_zero_kernel(float* __restrict__ out, int n4) {
    int i = blockIdx.x * blockDim.x + threadIdx.x;
    if (i < n4) {
        reinterpret_cast<float4*>(out)[i] = float4{0.f, 0.f, 0.f, 0.f};
    }
}

__global__ void __launch_bounds__(THREADS)
mp_scatter_add_kernel(const float* __restrict__ x,
                      const long long* __restrict__ src,
                      const long long* __restrict__ dst,
                      float* __restrict__ out,
                      int n_edges) {
    // s_idx[0..15] = src indices of this block's edges, s_idx[16..31] = dst.
    __shared__ long long s_idx[2 * EDGES_PER_BLOCK];

    const int t  = threadIdx.x;
    const int e0 = blockIdx.x * EDGES_PER_BLOCK;

    // Run-ahead prefetch of the index stream (global_prefetch_b8); speculative
    // OOB prefetches are silently dropped by hardware.
    if (t == 0) {
        __builtin_prefetch(&src[e0 + 16 * EDGES_PER_BLOCK], 0, 0);
        __builtin_prefetch(&dst[e0 + 16 * EDGES_PER_BLOCK], 0, 0);
    }

    const bool full_block = (e0 + EDGES_PER_BLOCK) <= n_edges;
    if (full_block) {
        // Async-copy 256B of indices into LDS: lanes 0..7 stage the 16 src
        // int64s (8 x b128), lanes 8..15 stage the 16 dst int64s.
        if (t < 16) {
            const long long* g = (t < 8) ? (src + e0 + 2 * t)
                                         : (dst + e0 + 2 * (t - 8));
            // generic->LDS address: low 32 bits of the flat shared address.
            unsigned lds_addr = (unsigned)(uintptr_t)(&s_idx[2 * t]);
            unsigned long long gaddr = (unsigned long long)(uintptr_t)g;
            asm volatile("global_load_async_to_lds_b128 %0, %1, off"
                         :: "v"(lds_addr), "v"(gaddr)
                         : "memory");
        }
    } else {
        // Tail block: guarded scalar staging.
        if (t < 2 * EDGES_PER_BLOCK) {
            int ee = e0 + (t & 15);
            if (ee < n_edges) {
                s_idx[t] = (t < EDGES_PER_BLOCK) ? src[ee] : dst[ee];
            }
        }
    }
    // Wave that issued the async copy drains ASYNCcnt; others pass trivially.
    asm volatile("s_wait_asynccnt 0x0" ::: "memory");
    __syncthreads();

    const int le = t >> 4;              // local edge 0..15
    const int e  = e0 + le;
    const int f  = (t & 15) << 2;       // feature offset 0,4,...,60
    if (e >= n_edges) return;

    // Broadcast reads from LDS (ds_load_b64, same address per 16-lane group).
    const long long s = s_idx[le];
    const long long d = s_idx[EDGES_PER_BLOCK + le];

    // Coalesced 256B row gather: 16 lanes each load float4 (global_load_b128).
    const float4 v = *reinterpret_cast<const float4*>(
        x + (size_t)s * DIM + (size_t)f);

    // Non-returning, relaxed, device-scope f32 atomics -> global_atomic_add_f32
    // (no return value: STOREcnt only, RMW resolved at L2).
    float* o = out + (size_t)d * DIM + (size_t)f;
    __hip_atomic_fetch_add(o + 0, v.x, __ATOMIC_RELAXED, __HIP_MEMORY_SCOPE_AGENT);
    __hip_atomic_fetch_add(o + 1, v.y, __ATOMIC_RELAXED, __HIP_MEMORY_SCOPE_AGENT);
    __hip_atomic_fetch_add(o + 2, v.z, __ATOMIC_RELAXED, __HIP_MEMORY_SCOPE_AGENT);
    __hip_atomic_fetch_add(o + 3, v.w, __ATOMIC_RELAXED, __HIP_MEMORY_SCOPE_AGENT);
}

extern "C" void kernel_launch(void* const* d_in, const int* in_sizes, int n_in,
                              void* d_out, int out_size, void* d_ws, size_t ws_size,
                              hipStream_t stream) {
    const float*     x   = (const float*)d_in[0];
    const long long* ei  = (const long long*)d_in[1];   // [2, E] row-major int64
    float*           out = (float*)d_out;

    const int n_edges = in_sizes[1] / 2;

    // 1) Zero the output (harness poisons d_out; segment_sum starts from 0).
    {
        const int n4     = out_size / 4;      // 64-dim rows -> divisible by 4
        const int blocks = (n4 + THREADS - 1) / THREADS;
        mp_zero_kernel<<<blocks, THREADS, 0, stream>>>(out, n4);
    }

    // 2) Scatter-add: 16 edges per 256-thread block, 16 lanes x float4 per edge.
    {
        const int blocks = (n_edges + EDGES_PER_BLOCK - 1) / EDGES_PER_BLOCK;
        mp_scatter_add_kernel<<<blocks, THREADS, 0, stream>>>(
            x, ei, ei + n_edges, out, n_edges);
    }
}